// GCNEncoder_27779848471424
// MI455X (gfx1250) — compile-verified
//
#include <hip/hip_runtime.h>
#include <math.h>
#include <stdint.h>

typedef __attribute__((ext_vector_type(2))) float v2f;
typedef __attribute__((ext_vector_type(8))) float v8f;

#define D 128

// ---------------------------------------------------------------------------
// GEMM: out[M x 128] = A[M x 128] @ W[128 x 128] (+ optional bias)
// One wave32 computes a 16-row x 128-col tile with V_WMMA_F32_16X16X4_F32.
// ---------------------------------------------------------------------------
__global__ __launch_bounds__(256) void gcn_gemm_f32_wmma(
    const float* __restrict__ A, const float* __restrict__ W,
    const float* __restrict__ bias, float* __restrict__ out,
    int rowTiles, int nRows)
{
  const int wave = threadIdx.x >> 5;
  const int lane = threadIdx.x & 31;
  const int tile = blockIdx.x * (blockDim.x >> 5) + wave;
  if (tile >= rowTiles) return;          // wave-uniform: EXEC stays all-ones
  const int m0   = tile * 16;
  const int half = lane >> 4;
  const int l    = lane & 15;

  v8f acc[8];
#pragma unroll
  for (int nt = 0; nt < 8; ++nt) acc[nt] = v8f{};

  int rowL = m0 + l;
  if (rowL >= nRows) rowL = nRows - 1;   // safety clamp (N % 16 == 0 normally)
  const float* __restrict__ arow = A + (size_t)rowL * D;

  for (int k = 0; k < D; k += 4) {
    const int ka = k + 2 * half;
    v2f a;
    a.x = arow[ka];
    a.y = arow[ka + 1];
    const float* __restrict__ w0 = W + (size_t)ka * D;
    const float* __restrict__ w1 = w0 + D;
#pragma unroll
    for (int nt = 0; nt < 8; ++nt) {
      const int n = nt * 16 + l;
      v2f b;
      b.x = w0[n];
      b.y = w1[n];
      acc[nt] = __builtin_amdgcn_wmma_f32_16x16x4_f32(
          false, a, false, b, (short)0, acc[nt], false, false);
    }
  }

#pragma unroll
  for (int nt = 0; nt < 8; ++nt) {
    const int n = nt * 16 + l;
    const float bb = bias ? bias[n] : 0.0f;
#pragma unroll
    for (int r = 0; r < 8; ++r) {
      const int row = m0 + r + 8 * half;
      if (row < nRows)
        out[(size_t)row * D + n] = acc[nt][r] + bb;
    }
  }
}

// Per-edge: weighted degree (for norm) and in-edge count (for CSR), one pass.
__global__ __launch_bounds__(256) void deg_count_kernel(
    const int* __restrict__ dst, const float* __restrict__ ew,
    float* __restrict__ deg, int* __restrict__ cnt, int E)
{
  int e = blockIdx.x * blockDim.x + threadIdx.x;
  if (e < E) {
    const int d = dst[e];
    atomicAdd(&deg[d], ew[e]);
    atomicAdd(&cnt[d], 1);
  }
}

// dinv = rsqrt(deg + 1)   (self-loop weight 1; deg+1 > 0 always)
__global__ __launch_bounds__(256) void dinv_kernel(
    const float* __restrict__ deg, float* __restrict__ dinv, int n)
{
  int i = blockIdx.x * blockDim.x + threadIdx.x;
  if (i < n) dinv[i] = rsqrtf(deg[i] + 1.0f);
}

// Single-workgroup exclusive scan: rowstart[0..n] from counts[0..n-1].
__global__ __launch_bounds__(1024) void scan_kernel(
    const int* __restrict__ counts, int* __restrict__ rowstart, int n)
{
  __shared__ int smem[1024];
  __shared__ int carry;
  if (threadIdx.x == 0) carry = 0;
  __syncthreads();
  for (int base = 0; base < n; base += 1024) {
    const int i = base + (int)threadIdx.x;
    const int v = (i < n) ? counts[i] : 0;
    smem[threadIdx.x] = v;
    __syncthreads();
    for (int off = 1; off < 1024; off <<= 1) {
      int t = 0;
      if ((int)threadIdx.x >= off) t = smem[threadIdx.x - off];
      __syncthreads();
      smem[threadIdx.x] += t;
      __syncthreads();
    }
    const int blockTotal = smem[1023];
    const int myIncl = smem[threadIdx.x];
    if (i < n) rowstart[i] = carry + myIncl - v;   // exclusive
    __syncthreads();                                // all reads of carry done
    if (threadIdx.x == 0) carry += blockTotal;
    __syncthreads();
  }
  if (threadIdx.x == 0) rowstart[n] = carry;
}

// Build CSR edge records: erec[pos] = (src, norm) with norm = dinv[s]*ew*dinv[d].
__global__ __launch_bounds__(256) void fill_kernel(
    const int* __restrict__ src, const int* __restrict__ dst,
    const float* __restrict__ ew, const float* __restrict__ dinv,
    const int* __restrict__ rowstart, int* __restrict__ cursor,
    int2* __restrict__ erec, int E)
{
  int e = blockIdx.x * blockDim.x + threadIdx.x;
  if (e >= E) return;
  const int s = src[e];
  const int d = dst[e];
  const float norm = dinv[s] * ew[e] * dinv[d];
  const int pos = rowstart[d] + atomicAdd(&cursor[d], 1);
  erec[pos] = make_int2(s, __float_as_int(norm));
}

// Gather-based aggregation, fused with self-loop + bias + ReLU epilogue.
// One wave per dst node; lane holds 4 contiguous cols in a float4 accumulator.
// out must NOT alias xw.
__global__ __launch_bounds__(256) void gather_agg_kernel(
    const float* __restrict__ xw, const int2* __restrict__ erec,
    const int* __restrict__ rowstart, const float* __restrict__ dinv,
    const float* __restrict__ bias, float* __restrict__ out,
    int doRelu, int n)
{
  const int lane = threadIdx.x & 31;
  const int node = (blockIdx.x * blockDim.x + threadIdx.x) >> 5;
  if (node >= n) return;                 // wave-uniform
  const int beg = rowstart[node];
  const int end = rowstart[node + 1];

  float4 acc = make_float4(0.f, 0.f, 0.f, 0.f);
  for (int j = beg; j < end; ++j) {
    const int2 r = erec[j];
    if (j + 1 < end) {                   // prefetch next source row (global_prefetch_b8)
      const int2 r2 = erec[j + 1];
      __builtin_prefetch(xw + (size_t)r2.x * D + lane * 4, 0, 3);
    }
    const float nrm = __int_as_float(r.y);
    const float4 v = ((const float4*)(xw + (size_t)r.x * D))[lane];
    acc.x = fmaf(v.x, nrm, acc.x);
    acc.y = fmaf(v.y, nrm, acc.y);
    acc.z = fmaf(v.z, nrm, acc.z);
    acc.w = fmaf(v.w, nrm, acc.w);
  }

  const float di = dinv[node];
  const float d2 = di * di;
  const float4 xs = ((const float4*)(xw + (size_t)node * D))[lane];
  const float4 b  = ((const float4*)bias)[lane];
  acc.x = fmaf(xs.x, d2, acc.x) + b.x;
  acc.y = fmaf(xs.y, d2, acc.y) + b.y;
  acc.z = fmaf(xs.z, d2, acc.z) + b.z;
  acc.w = fmaf(xs.w, d2, acc.w) + b.w;
  if (doRelu) {
    acc.x = fmaxf(acc.x, 0.f); acc.y = fmaxf(acc.y, 0.f);
    acc.z = fmaxf(acc.z, 0.f); acc.w = fmaxf(acc.w, 0.f);
  }
  ((float4*)(out + (size_t)node * D))[lane] = acc;
}

static inline char* align16(char* p) {
  return (char*)(((uintptr_t)p + 15) & ~(uintptr_t)15);
}

extern "C" void kernel_launch(void* const* d_in, const int* in_sizes, int n_in,
                              void* d_out, int out_size, void* d_ws, size_t ws_size,
                              hipStream_t stream)
{
  const float* x  = (const float*)d_in[0];
  const int*   ei = (const int*)  d_in[1];   // [2, E]
  const float* ew = (const float*)d_in[2];
  const float* W1 = (const float*)d_in[3];
  const float* b1 = (const float*)d_in[4];
  const float* W2 = (const float*)d_in[5];
  const float* b2 = (const float*)d_in[6];
  const float* Wp = (const float*)d_in[7];
  const float* bp = (const float*)d_in[8];

  const int Nn = in_sizes[0] / D;
  const int E  = in_sizes[2];
  const int* src = ei;
  const int* dst = ei + E;

  // workspace layout
  char* p = (char*)d_ws;
  float* deg      = (float*)p; p += (size_t)Nn * 4;
  float* dinv     = (float*)p; p += (size_t)Nn * 4;
  int*   rowstart = (int*)p;   p += (size_t)(Nn + 1) * 4;
  int*   cursor   = (int*)p;   p += (size_t)Nn * 4;
  p = align16(p);
  int2*  erec     = (int2*)p;  p += (size_t)E * 8;
  p = align16(p);
  float* bufA     = (float*)p; p += (size_t)Nn * D * 4;
  float* bufB     = (float*)p;
  float* out      = (float*)d_out;

  const int rowTiles   = (Nn + 15) / 16;       // 3125 (exact)
  const int gemmBlocks = (rowTiles + 7) / 8;   // 8 waves / block
  const int aggBlocks  = (Nn + 7) / 8;         // 8 node-waves / block

  // ---- build normalization + dst-CSR (once, reused by both layers) ----
  hipMemsetAsync(deg, 0, (size_t)Nn * 4, stream);
  hipMemsetAsync(cursor, 0, (size_t)Nn * 4, stream);
  deg_count_kernel<<<(E + 255) / 256, 256, 0, stream>>>(dst, ew, deg, cursor, E);
  dinv_kernel<<<(Nn + 255) / 256, 256, 0, stream>>>(deg, dinv, Nn);
  scan_kernel<<<1, 1024, 0, stream>>>(cursor, rowstart, Nn);
  hipMemsetAsync(cursor, 0, (size_t)Nn * 4, stream);
  fill_kernel<<<(E + 255) / 256, 256, 0, stream>>>(src, dst, ew, dinv, rowstart,
                                                   cursor, erec, E);

  // ---- layer 1 ----
  gcn_gemm_f32_wmma<<<gemmBlocks, 256, 0, stream>>>(x, W1, nullptr, bufA, rowTiles, Nn);
  gather_agg_kernel<<<aggBlocks, 256, 0, stream>>>(bufA, erec, rowstart, dinv, b1,
                                                   bufB, 1, Nn);

  // ---- layer 2 ----
  gcn_gemm_f32_wmma<<<gemmBlocks, 256, 0, stream>>>(bufB, W2, nullptr, bufA, rowTiles, Nn);
  gather_agg_kernel<<<aggBlocks, 256, 0, stream>>>(bufA, erec, rowstart, dinv, b2,
                                                   bufB, 1, Nn);

  // ---- projection (bias fused, no relu) ----
  gcn_gemm_f32_wmma<<<gemmBlocks, 256, 0, stream>>>(bufB, Wp, bp, out, rowTiles, Nn);
}